// GAT_22514218566434
// MI455X (gfx1250) — compile-verified
//
#include <hip/hip_runtime.h>
#include <hip/hip_bf16.h>
#include <math.h>

typedef __attribute__((ext_vector_type(2))) float v2f;
typedef __attribute__((ext_vector_type(8))) float v8f;

#define HEADS 8
#define HCDIM 64
#define NEG_SLOPE 0.2f

// ---------- float <-> order-preserving uint key (for atomic segment-max) ----
__device__ __forceinline__ unsigned f2key(float f) {
    unsigned b = __float_as_uint(f);
    return (b & 0x80000000u) ? ~b : (b ^ 0x80000000u);
}
__device__ __forceinline__ float key2f(unsigned u) {
    unsigned b = (u & 0x80000000u) ? (u ^ 0x80000000u) : ~u;
    return __uint_as_float(b);
}

__device__ __forceinline__ void edge_sd(const int* __restrict__ ei, int E, int e,
                                        int& s, int& d) {
    if (e < E) { s = ei[e]; d = ei[E + e]; }
    else       { s = e - E; d = e - E; }     // appended self-loops
}

// ---------------------------------------------------------------------------
// GEMM: C[M,Ncols] = A[M,K] @ B[K,Ncols], fp32 WMMA 16x16x4.
// One wave per 16x16 output tile; block = 128 threads = 4 waves = 4 col tiles
// (Ncols == 64). M multiple of 16, K multiple of 4 (exact for this problem).
// ---------------------------------------------------------------------------
__global__ void __launch_bounds__(128)
wmma_gemm_f32(const float* __restrict__ A, const float* __restrict__ B,
              float* __restrict__ C, int K, int Ncols) {
    const int wave = threadIdx.x >> 5;
    const int lane = threadIdx.x & 31;
    const int m0   = blockIdx.x * 16;
    const int n0   = wave * 16;
    const int half = lane >> 4;      // which half-wave
    const int r    = lane & 15;      // row within tile (A/B/C lane index)

    v8f acc = {};
    const float* Arow = A + (size_t)(m0 + r) * K;
    for (int k0 = 0; k0 < K; k0 += 4) {
        const int ka = k0 + 2 * half;
        v2f a, b;
        // A 16x4 frag: lanes0-15 K={k0,k0+1}, lanes16-31 K={k0+2,k0+3}
        a.x = Arow[ka];
        a.y = Arow[ka + 1];
        // B 4x16 frag: row striped across lanes within each VGPR
        b.x = B[(size_t)ka       * Ncols + n0 + r];
        b.y = B[(size_t)(ka + 1) * Ncols + n0 + r];
        acc = __builtin_amdgcn_wmma_f32_16x16x4_f32(
            false, a, false, b, (short)0, acc, false, false);
    }
    // C/D layout: VGPR v -> M = v (lanes 0-15) / v+8 (lanes 16-31), N = r
#pragma unroll
    for (int v = 0; v < 8; ++v) {
        C[(size_t)(m0 + v + 8 * half) * Ncols + n0 + r] = acc[v];
    }
}

// ---------------------------------------------------------------------------
// alpha_src / alpha_dst: per (node, head) dot of h[n, hd*8:(hd+1)*8] with a_*
// ---------------------------------------------------------------------------
__global__ void alpha_kernel(const float* __restrict__ h,
                             const float* __restrict__ a_s,
                             const float* __restrict__ a_d,
                             float* __restrict__ as, float* __restrict__ ad,
                             int Nn) {
    int t = blockIdx.x * blockDim.x + threadIdx.x;
    if (t >= Nn * HEADS) return;
    int hd = t & (HEADS - 1);
    int n  = t >> 3;
    const float* hp = h + (size_t)n * HCDIM + hd * 8;
    float ss = 0.f, sd = 0.f;
#pragma unroll
    for (int c = 0; c < 8; ++c) {
        ss += hp[c] * a_s[hd * 8 + c];
        sd += hp[c] * a_d[hd * 8 + c];
    }
    as[t] = ss;
    ad[t] = sd;
}

// ---------------------------------------------------------------------------
// pass 1: per (edge, head) leaky-relu score -> atomic segment-max (uint keys)
// ---------------------------------------------------------------------------
__global__ void edge_max_kernel(const int* __restrict__ ei, int E, int Nn,
                                const float* __restrict__ as,
                                const float* __restrict__ ad,
                                unsigned* __restrict__ mkey) {
    int t = blockIdx.x * blockDim.x + threadIdx.x;
    int total = (E + Nn) * HEADS;
    if (t >= total) return;
    int hd = t & (HEADS - 1);
    int e  = t >> 3;
    int s, d; edge_sd(ei, E, e, s, d);
    float ev = as[s * HEADS + hd] + ad[d * HEADS + hd];
    ev = ev > 0.f ? ev : NEG_SLOPE * ev;
    atomicMax(&mkey[d * HEADS + hd], f2key(ev));
}

// ---------------------------------------------------------------------------
// pass 2: per (edge, head) exp(e - m[dst]) -> atomic segment-sum
// ---------------------------------------------------------------------------
__global__ void edge_sum_kernel(const int* __restrict__ ei, int E, int Nn,
                                const float* __restrict__ as,
                                const float* __restrict__ ad,
                                const unsigned* __restrict__ mkey,
                                float* __restrict__ ssum) {
    int t = blockIdx.x * blockDim.x + threadIdx.x;
    int total = (E + Nn) * HEADS;
    if (t >= total) return;
    int hd = t & (HEADS - 1);
    int e  = t >> 3;
    int s, d; edge_sd(ei, E, e, s, d);
    float ev = as[s * HEADS + hd] + ad[d * HEADS + hd];
    ev = ev > 0.f ? ev : NEG_SLOPE * ev;
    float m = key2f(mkey[d * HEADS + hd]);
    atomicAdd(&ssum[d * HEADS + hd], __expf(ev - m));
}

// ---------------------------------------------------------------------------
// pass 3: one wave32 per edge; lane handles channels {lane, lane+32};
// agg[dst, c] += h[src, c] * alpha(edge, head(c))
// ---------------------------------------------------------------------------
__global__ void __launch_bounds__(256)
edge_agg_kernel(const int* __restrict__ ei, int E, int Nn,
                const float* __restrict__ h,
                const float* __restrict__ as, const float* __restrict__ ad,
                const unsigned* __restrict__ mkey,
                const float* __restrict__ ssum,
                float* __restrict__ agg) {
    int lane = threadIdx.x & 31;
    int wid  = blockIdx.x * (blockDim.x >> 5) + (threadIdx.x >> 5);
    if (wid >= E + Nn) return;
    int s, d; edge_sd(ei, E, wid, s, d);
#pragma unroll
    for (int half = 0; half < 2; ++half) {
        int c  = lane + 32 * half;
        int hd = c >> 3;
        float ev = as[s * HEADS + hd] + ad[d * HEADS + hd];
        ev = ev > 0.f ? ev : NEG_SLOPE * ev;
        float m = key2f(mkey[d * HEADS + hd]);
        float a = __expf(ev - m) / (ssum[d * HEADS + hd] + 1e-16f);
        atomicAdd(&agg[(size_t)d * HCDIM + c], h[(size_t)s * HCDIM + c] * a);
    }
}

// ---------------------------------------------------------------------------
// out = elu(agg + bias)
// ---------------------------------------------------------------------------
__global__ void bias_elu_kernel(const float* __restrict__ agg,
                                const float* __restrict__ b,
                                float* __restrict__ out, int total) {
    int t = blockIdx.x * blockDim.x + threadIdx.x;
    if (t >= total) return;
    float v = agg[t] + b[t & (HCDIM - 1)];
    out[t] = v > 0.f ? v : (__expf(v) - 1.0f);
}

// ---------------------------------------------------------------------------
// global mean pool: block per node, 64 threads (one channel each)
// ---------------------------------------------------------------------------
__global__ void pool_kernel(const float* __restrict__ h,
                            const int* __restrict__ batch,
                            float* __restrict__ sums, float* __restrict__ cnt,
                            int Nn) {
    int n = blockIdx.x;
    int c = threadIdx.x;
    if (n >= Nn) return;
    int g = batch[n];
    atomicAdd(&sums[(size_t)g * HCDIM + c], h[(size_t)n * HCDIM + c]);
    if (c == 0) atomicAdd(&cnt[g], 1.0f);
}

// ---------------------------------------------------------------------------
// final: pooled @ Wlin + blin, log_softmax (2 classes), one thread per graph
// ---------------------------------------------------------------------------
__global__ void final_kernel(const float* __restrict__ sums,
                             const float* __restrict__ cnt,
                             const float* __restrict__ Wlin,
                             const float* __restrict__ blin,
                             float* __restrict__ out, int G) {
    int g = threadIdx.x;
    if (g >= G) return;
    float c  = fmaxf(cnt[g], 1.0f);
    float l0 = blin[0], l1 = blin[1];
#pragma unroll 8
    for (int k = 0; k < HCDIM; ++k) {
        float p = sums[(size_t)g * HCDIM + k] / c;
        l0 += p * Wlin[k * 2 + 0];
        l1 += p * Wlin[k * 2 + 1];
    }
    float mx  = fmaxf(l0, l1);
    float lse = mx + logf(__expf(l0 - mx) + __expf(l1 - mx));
    out[g * 2 + 0] = l0 - lse;
    out[g * 2 + 1] = l1 - lse;
}

// ---------------------------------------------------------------------------
extern "C" void kernel_launch(void* const* d_in, const int* in_sizes, int n_in,
                              void* d_out, int out_size, void* d_ws, size_t ws_size,
                              hipStream_t stream) {
    const float* x     = (const float*)d_in[0];
    const int*   ei    = (const int*)d_in[1];
    const int*   batch = (const int*)d_in[2];
    const float* W[3]  = {(const float*)d_in[3], (const float*)d_in[7],  (const float*)d_in[11]};
    const float* Asw[3]= {(const float*)d_in[4], (const float*)d_in[8],  (const float*)d_in[12]};
    const float* Adw[3]= {(const float*)d_in[5], (const float*)d_in[9],  (const float*)d_in[13]};
    const float* Bw[3] = {(const float*)d_in[6], (const float*)d_in[10], (const float*)d_in[14]};
    const float* Wlin  = (const float*)d_in[15];
    const float* blin  = (const float*)d_in[16];

    const int F_IN = 160;
    const int Nn = in_sizes[0] / F_IN;   // 50000
    const int E  = in_sizes[1] / 2;      // 1600000
    const int G  = 64;
    const int Et = E + Nn;               // with self-loops

    // workspace carve-out (fp32 everywhere)
    char* w = (char*)d_ws;
    float*    hbuf  = (float*)w;    w += (size_t)Nn * HCDIM * 4;
    float*    aggb  = (float*)w;    w += (size_t)Nn * HCDIM * 4;
    float*    inb   = (float*)w;    w += (size_t)Nn * HCDIM * 4;
    float*    as    = (float*)w;    w += (size_t)Nn * HEADS * 4;
    float*    ad    = (float*)w;    w += (size_t)Nn * HEADS * 4;
    unsigned* mkey  = (unsigned*)w; w += (size_t)Nn * HEADS * 4;
    float*    ssum  = (float*)w;    w += (size_t)Nn * HEADS * 4;
    float*    sums  = (float*)w;    w += (size_t)G * HCDIM * 4;
    float*    cnt   = (float*)w;    w += (size_t)G * 4;

    const int blk = 256;
    const int nhBlocks   = (Nn * HEADS + blk - 1) / blk;
    const int ehBlocks   = (Et * HEADS + blk - 1) / blk;
    const int aggBlocks  = (Et + 7) / 8;            // 8 waves per 256-thread block
    const int elemBlocks = (Nn * HCDIM + blk - 1) / blk;

    const float* layer_in = x;
    for (int L = 0; L < 3; ++L) {
        const int K = (L == 0) ? F_IN : HCDIM;
        // 1) projection GEMM via fp32 WMMA
        wmma_gemm_f32<<<Nn / 16, 128, 0, stream>>>(layer_in, W[L], hbuf, K, HCDIM);
        // 2) per-node per-head attention logits
        alpha_kernel<<<nhBlocks, blk, 0, stream>>>(hbuf, Asw[L], Adw[L], as, ad, Nn);
        // 3) segment-max over incoming edges
        hipMemsetAsync(mkey, 0, (size_t)Nn * HEADS * 4, stream);
        edge_max_kernel<<<ehBlocks, blk, 0, stream>>>(ei, E, Nn, as, ad, mkey);
        // 4) segment-sum of exp(e - m)
        hipMemsetAsync(ssum, 0, (size_t)Nn * HEADS * 4, stream);
        edge_sum_kernel<<<ehBlocks, blk, 0, stream>>>(ei, E, Nn, as, ad, mkey, ssum);
        // 5) normalized scatter-aggregate
        hipMemsetAsync(aggb, 0, (size_t)Nn * HCDIM * 4, stream);
        edge_agg_kernel<<<aggBlocks, blk, 0, stream>>>(ei, E, Nn, hbuf, as, ad,
                                                       mkey, ssum, aggb);
        // 6) bias + ELU -> next layer input
        bias_elu_kernel<<<elemBlocks, blk, 0, stream>>>(aggb, Bw[L], inb, Nn * HCDIM);
        layer_in = inb;
    }

    // global mean pool + classifier head
    hipMemsetAsync(sums, 0, (size_t)G * HCDIM * 4, stream);
    hipMemsetAsync(cnt, 0, (size_t)G * 4, stream);
    pool_kernel<<<Nn, HCDIM, 0, stream>>>(inb, batch, sums, cnt, Nn);
    final_kernel<<<1, G, 0, stream>>>(sums, cnt, Wlin, blin, (float*)d_out, G);
}